// Helix_59442347377535
// MI455X (gfx1250) — compile-verified
//
#include <hip/hip_runtime.h>
#include <math.h>

typedef __attribute__((ext_vector_type(2))) float v2f;
typedef __attribute__((ext_vector_type(4))) float v4f;
typedef __attribute__((ext_vector_type(8))) float v8f;

#define WAVES_PER_BLOCK 8
// per wave: 96 floats positions + 288 floats rotations + 8 floats dump/pad = 392
#define WAVE_STAGE_FLOATS 392

// Reference's exact f32 angle arithmetic (no fp-contract), given (h, s) directly.
__device__ __forceinline__ float angle_from_hs(int h, int s, float nstartF,
                                               float twist_rad, bool wantB) {
  // sym = f32(2*pi) * h / n_start   (matches 2.0*jnp.pi*arange/n_start in f32)
  float sym = __fdiv_rn(__fmul_rn(6.28318530717958647692f, (float)h), nstartF);
  float t = __fmul_rn((float)s, twist_rad);
  // b = k*twist - sym ; a = -(sym + k*twist)
  return wantB ? __fsub_rn(t, sym) : -__fadd_rn(sym, t);
}

__global__ void __launch_bounds__(256) Helix_59442347377535_kernel(
    const float* __restrict__ d_rise, const float* __restrict__ d_twist,
    const float* __restrict__ d_disp, const float* __restrict__ d_rot0,
    const int* __restrict__ d_nstart, const int* __restrict__ d_nsub,
    float* __restrict__ out, int total) {
  __shared__ __align__(16) float smem[WAVES_PER_BLOCK * WAVE_STAGE_FLOATS];

  const int tid  = threadIdx.x;
  const int lane = tid & 31;
  const int wib  = tid >> 5;
  const int base = blockIdx.x * (WAVES_PER_BLOCK * 32) + wib * 32;
  if (base >= total) return;  // wave-uniform exit; no barriers used below

  const int   S       = d_nsub[0];
  const int   nstart  = d_nstart[0];
  const float nstartF = (float)nstart;
  const float rise    = d_rise[0];
  // twist_rad = twist * f32(pi/180), single f32 multiply like jnp.deg2rad
  const float twist_rad = __fmul_rn(d_twist[0], 0.01745329251994329577f);
  const float x0 = d_disp[0], y0 = d_disp[1], z0 = d_disp[2];
  // rise * n_subunits / 2 with reference op order
  const float halfShift = __fmul_rn(__fmul_rn(rise, (float)S), 0.5f);

  // One wave-uniform division; per-lane (h,s) by incremental carry (o < 32 <= S).
  const int hb = base / S;
  const int sb = base - hb * S;

  const int col  = lane & 15;  // N column this lane feeds/receives
  const int half = lane >> 4;  // 0: supplies A cols 0,1 (cb,sb); 1: cols 2,3 (ca,sa)

  // ---------------- B operand: constant 4x16 f32 ----------------
  // rows K0,K1 (lanes 0-15): rotation combine; rows K2,K3 (lanes 16-31): position combine
  // cols 0..2 : D = cb*T0r0 - sb*T0r1   (rotation row 0)
  // cols 3..5 : D = sb*T0r0 + cb*T0r1   (rotation row 1)
  // col  6    : D = ca*x0 - sa*y0       (px)
  // col  7    : D = sa*x0 + ca*y0       (py)
  float bx = 0.f, by = 0.f;
  if (half == 0) {
    if (col < 3)      { bx =  d_rot0[0 * 3 + col];       by = -d_rot0[1 * 3 + col]; }
    else if (col < 6) { bx =  d_rot0[1 * 3 + (col - 3)]; by =  d_rot0[0 * 3 + (col - 3)]; }
  } else {
    if (col == 6)      { bx = x0; by = -y0; }
    else if (col == 7) { bx = y0; by =  x0; }
  }
  v2f Bm; Bm.x = bx; Bm.y = by;

  // ---------------- A operands: one sincos per lane per 16-element group ----------------
  const bool wantB = (half == 0);
  int o1 = col;      if (base + o1 >= total) o1 = total - 1 - base;
  int o2 = col + 16; if (base + o2 >= total) o2 = total - 1 - base;
  int s1 = sb + o1, h1 = hb; if (s1 >= S) { s1 -= S; h1 += 1; }
  int s2 = sb + o2, h2 = hb; if (s2 >= S) { s2 -= S; h2 += 1; }
  float a1 = angle_from_hs(h1, s1, nstartF, twist_rad, wantB);
  float a2 = angle_from_hs(h2, s2, nstartF, twist_rad, wantB);
  float sn1, cs1, sn2, cs2;
  sincosf(a1, &sn1, &cs1);  // ocml: full range reduction, matches XLA lowering
  sincosf(a2, &sn2, &cs2);
  v2f A1; A1.x = cs1; A1.y = sn1;
  v2f A2; A2.x = cs2; A2.y = sn2;

  v8f Cz = {0.f, 0.f, 0.f, 0.f, 0.f, 0.f, 0.f, 0.f};
  // D(16x16) = A(16x4) x B(4x16), f32: 8 useful columns per element row
  v8f D1 = __builtin_amdgcn_wmma_f32_16x16x4_f32(false, A1, false, Bm,
                                                 (short)0, Cz, false, false);
  v8f D2 = __builtin_amdgcn_wmma_f32_16x16x4_f32(false, A2, false, Bm,
                                                 (short)0, Cz, false, false);

  float* wstage = smem + wib * WAVE_STAGE_FLOATS;
  float* posS = wstage;        // 96 floats: 32 x (px,py,pz)
  float* rotS = wstage + 96;   // 288 floats: 32 x 9
                               // wstage[384]: dump dword for unused columns

  // ---------------- scatter D into element-major staging (branch-free loop) ----------------
  // D layout: VGPR j -> row m = half*8 + j, col n = lane&15.
  // Per-lane destination is a (base, stride) pair chosen once from `col`.
  int fbase, fstride;
  if (col < 6)      { fbase = 96 + (half * 8) * 9 + col;      fstride = 9; }
  else if (col < 8) { fbase = (half * 8) * 3 + (col - 6);     fstride = 3; }
  else              { fbase = 384;                            fstride = 0; }
  const int fskip = 16 * fstride;  // D2 elements sit 16 elements later
#pragma unroll
  for (int j = 0; j < 8; ++j) {
    wstage[fbase + j * fstride]         = D1[j];
    wstage[fbase + fskip + j * fstride] = D2[j];
  }

  // ---------------- fill pz and constant rotation row 2 (one element per lane) ----------------
  {
    int o = lane; if (base + o >= total) o = total - 1 - base;
    int s = sb + o; if (s >= S) s -= S;
    float pz = __fsub_rn(__fadd_rn(z0, __fmul_rn((float)s, rise)), halfShift);
    posS[lane * 3 + 2] = pz;
    rotS[lane * 9 + 6] = d_rot0[6];
    rotS[lane * 9 + 7] = d_rot0[7];
    rotS[lane * 9 + 8] = d_rot0[8];
  }

  float* __restrict__ posOut = out;
  float* __restrict__ rotOut = out + (size_t)total * 3;

  if (base + 32 <= total) {
    // Fully-coalesced streaming copy-out: 24 + 72 b128 non-temporal stores per wave.
    const v4f* ps4 = (const v4f*)posS;
    const v4f* rs4 = (const v4f*)rotS;
    v4f* pd4 = (v4f*)(posOut + (size_t)base * 3);
    v4f* rd4 = (v4f*)(rotOut + (size_t)base * 9);
    if (lane < 24) __builtin_nontemporal_store(ps4[lane], pd4 + lane);
#pragma unroll
    for (int p = 0; p < 3; ++p) {
      int c = p * 32 + lane;
      if (c < 72) __builtin_nontemporal_store(rs4[c], rd4 + c);
    }
  } else {
    // tail wave: per-element scalar stores
    int e = base + lane;
    if (e < total) {
      posOut[(size_t)e * 3 + 0] = posS[lane * 3 + 0];
      posOut[(size_t)e * 3 + 1] = posS[lane * 3 + 1];
      posOut[(size_t)e * 3 + 2] = posS[lane * 3 + 2];
#pragma unroll
      for (int c = 0; c < 9; ++c)
        rotOut[(size_t)e * 9 + c] = rotS[lane * 9 + c];
    }
  }
}

extern "C" void kernel_launch(void* const* d_in, const int* in_sizes, int n_in,
                              void* d_out, int out_size, void* d_ws, size_t ws_size,
                              hipStream_t stream) {
  (void)in_sizes; (void)n_in; (void)d_ws; (void)ws_size;
  const float* d_rise   = (const float*)d_in[0];
  const float* d_twist  = (const float*)d_in[1];
  const float* d_disp   = (const float*)d_in[2];
  const float* d_rot0   = (const float*)d_in[3];
  const int*   d_nstart = (const int*)d_in[4];
  const int*   d_nsub   = (const int*)d_in[5];

  int total = out_size / 12;  // 3 position + 9 rotation floats per subunit
  int blocks = (total + 255) / 256;
  hipLaunchKernelGGL(Helix_59442347377535_kernel, dim3(blocks), dim3(256), 0, stream,
                     d_rise, d_twist, d_disp, d_rot0, d_nstart, d_nsub,
                     (float*)d_out, total);
}